// MagNet_59674275611203
// MI455X (gfx1250) — compile-verified
//
#include <hip/hip_runtime.h>
#include <hip/hip_bf16.h>

#define NN 50000
#define EE 800000
#define K1 3

typedef __attribute__((ext_vector_type(2))) float v2f;
typedef __attribute__((ext_vector_type(8))) float v8f;

// ---------------------------------------------------------------------------
// Zero two buffers (T_real / T_imag)
__global__ void zero2_kernel(float* __restrict__ a, float* __restrict__ b, int n) {
    int i = blockIdx.x * blockDim.x + threadIdx.x;
    if (i < n) { a[i] = 0.f; b[i] = 0.f; }
}

// Initialize layer accumulators with the bias (Cout = 64)
__global__ void bias_init_kernel(float* __restrict__ r, float* __restrict__ im,
                                 const float* __restrict__ bias, int n) {
    int i = blockIdx.x * blockDim.x + threadIdx.x;
    if (i < n) { float bv = bias[i & 63]; r[i] = bv; im[i] = bv; }
}

// ---------------------------------------------------------------------------
// Fused complex SpMM for one Chebyshev term:
//   T_real[r] += vr*Xr[c] - vi*Xi[c];  T_imag[r] += vi*Xr[c] + vr*Xi[c]
// Edge-parallel, 4 channels per thread (float4 gathers), fp32 global atomics.
__global__ void spmm_kernel(const int* __restrict__ rows, const int* __restrict__ cols,
                            const float* __restrict__ vr, const float* __restrict__ vi,
                            const float* __restrict__ Xr, const float* __restrict__ Xi,
                            float* __restrict__ Tr, float* __restrict__ Ti, int Cin) {
    unsigned idx = blockIdx.x * blockDim.x + threadIdx.x;
    unsigned chunks = (unsigned)(Cin >> 2);
    unsigned e = idx / chunks;
    unsigned q = idx - e * chunks;
    if (e >= EE) return;
    int r = rows[e], c = cols[e];
    float a = vr[e], b = vi[e];
    const float4 xr = *(const float4*)(Xr + (size_t)c * Cin + q * 4);
    const float4 xi = *(const float4*)(Xi + (size_t)c * Cin + q * 4);
    float* tr = Tr + (size_t)r * Cin + q * 4;
    float* ti = Ti + (size_t)r * Cin + q * 4;
    atomicAdd(tr + 0, a * xr.x - b * xi.x);
    atomicAdd(tr + 1, a * xr.y - b * xi.y);
    atomicAdd(tr + 2, a * xr.z - b * xi.z);
    atomicAdd(tr + 3, a * xr.w - b * xi.w);
    atomicAdd(ti + 0, b * xr.x + a * xi.x);
    atomicAdd(ti + 1, b * xr.y + a * xi.y);
    atomicAdd(ti + 2, b * xr.z + a * xi.z);
    atomicAdd(ti + 3, b * xr.w + a * xi.w);
}

// ---------------------------------------------------------------------------
// WMMA fp32 GEMM-accumulate: out += T[rt*16.., :Cin] @ W[Cin, 64]
// grid = (3125, 2) blocks of 128 threads; blockIdx.y selects real/imag.
// Each of the 4 waves owns one 16x16 output tile; K stepped by 4 via
// V_WMMA_F32_16X16X4_F32. A-tile (16xCin) and W (Cinx64) staged in LDS.
__global__ __launch_bounds__(128) void gemm_wmma_kernel(
        const float* __restrict__ Tr, const float* __restrict__ Ti,
        const float* __restrict__ W,
        float* __restrict__ outR, float* __restrict__ outI, int Cin) {
    __shared__ float sT[16 * 128];   // up to 8 KB
    __shared__ float sW[128 * 64];   // up to 32 KB
    const float* T = blockIdx.y ? Ti : Tr;
    float* out = blockIdx.y ? outI : outR;
    const int rt = blockIdx.x;

    for (int idx = threadIdx.x; idx < 16 * Cin; idx += 128)
        sT[idx] = T[(size_t)rt * 16 * Cin + idx];           // coalesced
    for (int idx = threadIdx.x; idx < Cin * 64; idx += 128)
        sW[idx] = W[idx];
    __syncthreads();

    const int lane = threadIdx.x & 31;
    const int ct   = threadIdx.x >> 5;   // column tile 0..3
    const int half = lane >> 4;          // upper/lower half of wave
    const int l15  = lane & 15;

    float* outTile = out + (size_t)rt * 16 * 64 + ct * 16;
    v8f c;
    // C/D layout: lane = col, VGPR v -> row = half*8 + v
#pragma unroll
    for (int v = 0; v < 8; ++v) c[v] = outTile[(half * 8 + v) * 64 + l15];

    for (int k0 = 0; k0 < Cin; k0 += 4) {
        v2f a, b;
        // A 16x4 layout: lane -> row = l15, VGPR0/1 -> K = half*2 + {0,1}
        a.x = sT[l15 * Cin + k0 + half * 2 + 0];
        a.y = sT[l15 * Cin + k0 + half * 2 + 1];
        // B 4x16 layout: lane -> col = l15, VGPR0/1 -> K = half*2 + {0,1}
        b.x = sW[(k0 + half * 2 + 0) * 64 + ct * 16 + l15];
        b.y = sW[(k0 + half * 2 + 1) * 64 + ct * 16 + l15];
        c = __builtin_amdgcn_wmma_f32_16x16x4_f32(
                /*neg_a=*/false, a, /*neg_b=*/false, b,
                /*c_mod=*/(short)0, c, /*reuse_a=*/false, /*reuse_b=*/false);
    }
#pragma unroll
    for (int v = 0; v < 8; ++v) outTile[(half * 8 + v) * 64 + l15] = c[v];
}

// ---------------------------------------------------------------------------
// Complex ReLU gated on the real part (in place)
__global__ void crelu_kernel(float* __restrict__ r, float* __restrict__ im, int n) {
    int i = blockIdx.x * blockDim.x + threadIdx.x;
    if (i < n) { float m = (r[i] >= 0.f) ? 1.f : 0.f; r[i] *= m; im[i] *= m; }
}

// Concatenate [N,64]+[N,64] -> x [N,128]
__global__ void concat_kernel(const float* __restrict__ r, const float* __restrict__ im,
                              float* __restrict__ x) {
    int i = blockIdx.x * blockDim.x + threadIdx.x;
    if (i < NN * 128) {
        int node = i >> 7, f = i & 127;
        x[i] = (f < 64) ? r[(size_t)node * 64 + f] : im[(size_t)node * 64 + f - 64];
    }
}

// Per-channel column sums of x (for the avg-pool)
__global__ void colsum_kernel(const float* __restrict__ x, float* __restrict__ chansum) {
    __shared__ float red[256];
    const int c = blockIdx.x;
    float s = 0.f;
    for (int i = threadIdx.x; i < NN; i += 256) s += x[(size_t)i * 128 + c];
    red[threadIdx.x] = s;
    __syncthreads();
    for (int st = 128; st > 0; st >>= 1) {
        if (threadIdx.x < st) red[threadIdx.x] += red[threadIdx.x + st];
        __syncthreads();
    }
    if (threadIdx.x == 0) chansum[c] = red[0];
}

// Channel attention MLP (128->8->128 + sigmoid) and fold ca into classifier:
// wca[o,c] = conv_w[o,c] * ca[c].  Single block of 128 threads.
__global__ void ca_kernel(const float* __restrict__ chansum,
                          const float* __restrict__ ca_w1, const float* __restrict__ ca_w2,
                          const float* __restrict__ conv_w,
                          float* __restrict__ ca, float* __restrict__ wca) {
    __shared__ float avg[128];
    __shared__ float hid[8];
    __shared__ float cas[128];
    const int t = threadIdx.x;
    avg[t] = chansum[t] * (1.f / (float)NN);
    __syncthreads();
    if (t < 8) {
        float s = 0.f;
        for (int c = 0; c < 128; ++c) s += ca_w1[t * 128 + c] * avg[c];
        hid[t] = s > 0.f ? s : 0.f;
    }
    __syncthreads();
    float s = 0.f;
    for (int r = 0; r < 8; ++r) s += ca_w2[t * 8 + r] * hid[r];
    float cav = 1.f / (1.f + __expf(-s));
    ca[t] = cav;
    cas[t] = cav;
    __syncthreads();
    for (int idx = t; idx < 1280; idx += 128) wca[idx] = conv_w[idx] * cas[idx & 127];
}

// Spatial-attention inputs: per-node mean/max over channels of x*ca
__global__ void smeanmax_kernel(const float* __restrict__ x, const float* __restrict__ ca,
                                float* __restrict__ s_mean, float* __restrict__ s_max) {
    __shared__ float cas[128];
    if (threadIdx.x < 128) cas[threadIdx.x] = ca[threadIdx.x];
    __syncthreads();
    int i = blockIdx.x * 256 + threadIdx.x;
    if (i >= NN) return;
    float sm = 0.f, mx = -3.4e38f;
    for (int c = 0; c < 128; ++c) {
        float v = x[(size_t)i * 128 + c] * cas[c];
        sm += v;
        mx = v > mx ? v : mx;
    }
    s_mean[i] = sm * (1.f / 128.f);
    s_max[i] = mx;
}

// 7-tap spatial conv + sigmoid, then 1x1 conv (ca pre-folded) + log_softmax
__global__ void final_kernel(const float* __restrict__ x,
                             const float* __restrict__ s_mean, const float* __restrict__ s_max,
                             const float* __restrict__ sa_w, const float* __restrict__ wca,
                             const float* __restrict__ conv_b, float* __restrict__ out) {
    __shared__ float swca[1280];
    __shared__ float scb[10];
    __shared__ float ssw[14];
    for (int idx = threadIdx.x; idx < 1280; idx += 256) swca[idx] = wca[idx];
    if (threadIdx.x < 10) scb[threadIdx.x] = conv_b[threadIdx.x];
    if (threadIdx.x < 14) ssw[threadIdx.x] = sa_w[threadIdx.x];
    __syncthreads();
    int i = blockIdx.x * 256 + threadIdx.x;
    if (i >= NN) return;
    float conv = 0.f;
#pragma unroll
    for (int t = 0; t < 7; ++t) {
        int j = i + t - 3;
        if (j >= 0 && j < NN) conv += ssw[t] * s_mean[j] + ssw[7 + t] * s_max[j];
    }
    float sa = 1.f / (1.f + __expf(-conv));
    float dot[10];
#pragma unroll
    for (int o = 0; o < 10; ++o) dot[o] = 0.f;
    for (int c = 0; c < 128; ++c) {
        float xv = x[(size_t)i * 128 + c];
#pragma unroll
        for (int o = 0; o < 10; ++o) dot[o] += swca[o * 128 + c] * xv;
    }
    float lg[10], m = -3.4e38f;
#pragma unroll
    for (int o = 0; o < 10; ++o) { lg[o] = sa * dot[o] + scb[o]; m = lg[o] > m ? lg[o] : m; }
    float lse = 0.f;
#pragma unroll
    for (int o = 0; o < 10; ++o) lse += __expf(lg[o] - m);
    lse = __logf(lse);
#pragma unroll
    for (int o = 0; o < 10; ++o) out[(size_t)o * NN + i] = lg[o] - m - lse;
}

// ---------------------------------------------------------------------------
extern "C" void kernel_launch(void* const* d_in, const int* in_sizes, int n_in,
                              void* d_out, int out_size, void* d_ws, size_t ws_size,
                              hipStream_t stream) {
    const float* Xr     = (const float*)d_in[0];
    const float* Xi     = (const float*)d_in[1];
    const int*   rows   = (const int*)d_in[2];
    const int*   cols   = (const int*)d_in[3];
    const float* vr     = (const float*)d_in[4];
    const float* vi     = (const float*)d_in[5];
    const float* w0     = (const float*)d_in[6];
    const float* b0     = (const float*)d_in[7];
    const float* w1     = (const float*)d_in[8];
    const float* b1     = (const float*)d_in[9];
    const float* w2     = (const float*)d_in[10];
    const float* b2     = (const float*)d_in[11];
    const float* ca_w1  = (const float*)d_in[12];
    const float* ca_w2  = (const float*)d_in[13];
    const float* sa_w   = (const float*)d_in[14];
    const float* conv_w = (const float*)d_in[15];
    const float* conv_b = (const float*)d_in[16];
    float* out = (float*)d_out;

    // Workspace carve-up (floats)
    float* ws      = (float*)d_ws;
    float* Tr      = ws;                       // N*128
    float* Ti      = Tr + (size_t)NN * 128;    // N*128
    float* accA_r  = Ti + (size_t)NN * 128;    // N*64
    float* accA_i  = accA_r + (size_t)NN * 64;
    float* accB_r  = accA_i + (size_t)NN * 64;
    float* accB_i  = accB_r + (size_t)NN * 64;
    float* s_mean  = accB_i + (size_t)NN * 64; // N
    float* s_max   = s_mean + NN;              // N
    float* chansum = s_max + NN;               // 128
    float* ca      = chansum + 128;            // 128
    float* wca     = ca + 128;                 // 1280
    float* xcat    = Tr;                       // alias: T buffers free after layer 3

    const float* inR = Xr;
    const float* inI = Xi;
    int Cin = 128;

    for (int layer = 0; layer < 3; ++layer) {
        const float* W = (layer == 0) ? w0 : (layer == 1) ? w1 : w2;
        const float* B = (layer == 0) ? b0 : (layer == 1) ? b1 : b2;
        float* outR = (layer == 1) ? accB_r : accA_r;
        float* outI = (layer == 1) ? accB_i : accA_i;

        bias_init_kernel<<<(NN * 64 + 255) / 256, 256, 0, stream>>>(outR, outI, B, NN * 64);

        for (int k = 0; k < K1; ++k) {
            int nT = NN * Cin;
            zero2_kernel<<<(nT + 255) / 256, 256, 0, stream>>>(Tr, Ti, nT);

            unsigned threads = (unsigned)EE * (unsigned)(Cin >> 2);
            spmm_kernel<<<(threads + 255) / 256, 256, 0, stream>>>(
                rows + (size_t)k * EE, cols + (size_t)k * EE,
                vr + (size_t)k * EE, vi + (size_t)k * EE,
                inR, inI, Tr, Ti, Cin);

            gemm_wmma_kernel<<<dim3(NN / 16, 2), 128, 0, stream>>>(
                Tr, Ti, W + (size_t)k * Cin * 64, outR, outI, Cin);
        }

        crelu_kernel<<<(NN * 64 + 255) / 256, 256, 0, stream>>>(outR, outI, NN * 64);
        inR = outR;
        inI = outI;
        Cin = 64;
    }

    // inR/inI now point at the layer-3 output (accA)
    concat_kernel<<<(NN * 128 + 255) / 256, 256, 0, stream>>>(inR, inI, xcat);
    colsum_kernel<<<128, 256, 0, stream>>>(xcat, chansum);
    ca_kernel<<<1, 128, 0, stream>>>(chansum, ca_w1, ca_w2, conv_w, ca, wca);
    smeanmax_kernel<<<(NN + 255) / 256, 256, 0, stream>>>(xcat, ca, s_mean, s_max);
    final_kernel<<<(NN + 255) / 256, 256, 0, stream>>>(xcat, s_mean, s_max,
                                                       sa_w, wca, conv_b, out);
}